// InfoGraph_10325101380017
// MI455X (gfx1250) — compile-verified
//
#include <hip/hip_runtime.h>
#include <hip/hip_bf16.h>
#include <math.h>

#define DIM   64
#define F_IN  30
#define F_INP 32
#define EDIM  11
#define HID   128
#define NB    128         // num_graphs (fixed by harness setup)
#define G3    (3*DIM)     // 192
#define EWN   (DIM*DIM)   // 4096

typedef float v2f __attribute__((ext_vector_type(2)));
typedef float v8f __attribute__((ext_vector_type(8)));

__device__ __forceinline__ float sigmf(float x) { return 1.0f / (1.0f + expf(-x)); }

// ---------------------------------------------------------------------------
__global__ void k_fill(float* p, float v, long n) {
    long i = (long)blockIdx.x * blockDim.x + threadIdx.x;
    if (i < n) p[i] = v;
}

// Pad node features [N,30] -> [N,32]
__global__ void k_pad_nf(const float* __restrict__ nf, float* __restrict__ ap, int N) {
    int i = blockIdx.x * blockDim.x + threadIdx.x;
    if (i >= N * F_INP) return;
    int n = i / F_INP, k = i - n * F_INP;
    ap[i] = (k < F_IN) ? nf[n * F_IN + k] : 0.0f;
}

// Pack row-major B[K,N] (zero-padded to Kp rows) into WMMA-frag order:
// Bp[((k>>2)*2 + ((k&3)>>1))*N + n)*2 + (k&1)] = B[k,n]
// so a lane's (B[k+kh,n], B[k+kh+1,n]) pair is one contiguous b64 load.
__global__ void k_packB(const float* __restrict__ B, float* __restrict__ Bp,
                        int Ksrc, int Kp, int N) {
    int i = blockIdx.x * blockDim.x + threadIdx.x;
    if (i >= Kp * N) return;
    int k = i / N, n = i - k * N;
    float v = (k < Ksrc) ? B[(size_t)k * N + n] : 0.0f;
    int q = k >> 2, r = k & 3;
    Bp[(((size_t)q * 2 + (r >> 1)) * N + n) * 2 + (r & 1)] = v;
}

// Pack transposed weights: logical B[k,n] = w[n*cols + k]  (w:[rows,cols] -> K=cols, N=rows)
__global__ void k_packBT(const float* __restrict__ w, float* __restrict__ Bp,
                         int rows, int cols) {
    int i = blockIdx.x * blockDim.x + threadIdx.x;
    if (i >= rows * cols) return;
    int k = i / rows, n = i - k * rows;           // k < cols (=K), n < rows (=N)
    int q = k >> 2, r = k & 3;
    Bp[(((size_t)q * 2 + (r >> 1)) * rows + n) * 2 + (r & 1)] = w[(size_t)n * cols + k];
}

// ---------------------------------------------------------------------------
// WMMA fp32 GEMM: C[M,N] = A[M,K] @ B[K,N] + bias[N], optional ReLU.
// B is frag-packed (k_packB/k_packBT). One wave computes a 16 x (NT*16) strip
// (A-frag reused NT times; all frag loads are contiguous b64).
// Requires N % (NT*16) == 0, K % 4 == 0.
template <int NT, bool RELU>
__global__ void k_gemm_wmma(const float* __restrict__ A, const float* __restrict__ Bp,
                            float* __restrict__ C, const float* __restrict__ bias,
                            int M, int N, int K, int lda, int ldc) {
    const int wave   = blockIdx.x * (blockDim.x >> 5) + (threadIdx.x >> 5);
    const int lane   = threadIdx.x & 31;
    const int strips = N / (NT * 16);
    const int mt     = (M + 15) >> 4;
    if (wave >= mt * strips) return;
    const int sm  = (wave / strips) << 4;
    const int sn  = (wave % strips) * (NT * 16);
    const int row = lane & 15;
    const int hh  = lane >> 4;          // half-wave: A/B rows k+2hh, k+2hh+1

    v8f acc[NT] = {};
    int ar = sm + row; if (ar >= M) ar = M - 1;        // clamp: EXEC stays full
    const float* aptr = A + (size_t)ar * lda + 2 * hh;
    for (int k0 = 0; k0 < K; k0 += 4) {
        v2f a = *(const v2f*)aptr;                     // A[ar][k0+2hh .. +1]
        aptr += 4;
        const float* bp = Bp + ((size_t)((k0 >> 2) * 2 + hh) * N + sn + row) * 2;
#pragma unroll
        for (int t = 0; t < NT; ++t) {
            v2f b = *(const v2f*)(bp + t * 32);        // B[k0+2hh..+1][sn+t*16+row]
            acc[t] = __builtin_amdgcn_wmma_f32_16x16x4_f32(
                false, a, false, b, (short)0, acc[t], false, false);
        }
    }
    const int rbase = sm + (hh << 3);                  // lanes 16-31: rows +8..+15
#pragma unroll
    for (int t = 0; t < NT; ++t) {
        const int col = sn + t * 16 + row;
        const float bv = bias[col];
#pragma unroll
        for (int v = 0; v < 8; ++v) {
            int r = rbase + v;
            if (r < M) {
                float val = acc[t][v] + bv;
                if (RELU) val = fmaxf(val, 0.0f);
                C[(size_t)r * ldc + col] = val;
            }
        }
    }
}

// ---------------------------------------------------------------------------
// enn hidden: h1[E,128] = relu(ef[E,11] @ w1[11,128] + b1)
__global__ void k_enn_hidden(const float* __restrict__ ef, const float* __restrict__ w1,
                             const float* __restrict__ b1, float* __restrict__ h1, int E) {
    int i = blockIdx.x * blockDim.x + threadIdx.x;
    if (i >= E * HID) return;
    int e = i >> 7, j = i & 127;
    const float* x = ef + (size_t)e * EDIM;
    float acc = b1[j];
#pragma unroll
    for (int k = 0; k < EDIM; ++k) acc = fmaf(x[k], w1[k * HID + j], acc);
    h1[i] = fmaxf(acc, 0.0f);
}

__global__ void k_deg(const int* __restrict__ dst, float* __restrict__ deg, int E) {
    int e = blockIdx.x * blockDim.x + threadIdx.x;
    if (e < E) atomicAdd(&deg[dst[e]], 1.0f);
}

__global__ void k_invdeg(float* deg, int N) {
    int n = blockIdx.x * blockDim.x + threadIdx.x;
    if (n < N) deg[n] = 1.0f / fmaxf(deg[n], 1.0f);
}

// msg[e,:] = feat[src[e],:] @ ew[e]  scattered into agg[dst[e],:] (atomics).
// One wave per edge; lane owns columns (lane, lane+32): coalesced 256B rows.
__global__ void k_msg(const float* __restrict__ feat, const float* __restrict__ ew,
                      const int* __restrict__ src, const int* __restrict__ dst,
                      float* __restrict__ agg, int e0, int ecnt) {
    int w = blockIdx.x * (blockDim.x >> 5) + (threadIdx.x >> 5);
    if (w >= ecnt) return;
    int lane = threadIdx.x & 31;
    int e = e0 + w;
    int s = src[e], d = dst[e];
    const float* W = ew + (size_t)w * EWN;      // chunk-local row
    const float* f = feat + (size_t)s * DIM;
    float a0 = 0.0f, a1 = 0.0f;
#pragma unroll 4
    for (int i = 0; i < DIM; ++i) {
        float fv = f[i];
        __builtin_prefetch(W + (i + 8) * DIM + lane, 0, 0);
        a0 = fmaf(fv, W[i * DIM + lane], a0);
        a1 = fmaf(fv, W[i * DIM + lane + 32], a1);
    }
    atomicAdd(&agg[(size_t)d * DIM + lane], a0);
    atomicAdd(&agg[(size_t)d * DIM + lane + 32], a1);
}

// m = relu(agg * inv_deg + conv_b)
__global__ void k_mrelu(const float* __restrict__ agg, const float* __restrict__ invd,
                        const float* __restrict__ cb, float* __restrict__ m, int N) {
    int i = blockIdx.x * blockDim.x + threadIdx.x;
    if (i >= N * DIM) return;
    int n = i >> 6, o = i & 63;
    m[i] = fmaxf(fmaf(agg[i], invd[n], cb[o]), 0.0f);
}

// GRU elementwise from separated x/h gate pre-activations
__global__ void k_gru(const float* __restrict__ Gi, const float* __restrict__ Gh,
                      float* __restrict__ h, int N) {
    int i = blockIdx.x * blockDim.x + threadIdx.x;
    if (i >= N * DIM) return;
    int n = i >> 6, o = i & 63;
    const float* gi = Gi + (size_t)n * G3;
    const float* gh = Gh + (size_t)n * G3;
    float r  = sigmf(gi[o] + gh[o]);
    float z  = sigmf(gi[DIM + o] + gh[DIM + o]);
    float nn = tanhf(gi[2 * DIM + o] + r * gh[2 * DIM + o]);
    h[i] = (1.0f - z) * nn + z * h[i];
}

// Set2Set LSTM: one thread per (graph, unit); double-buffered h
__global__ void k_lstm(const float* __restrict__ q, const float* __restrict__ hin,
                       float* __restrict__ hout, float* __restrict__ c,
                       const float* __restrict__ wih, const float* __restrict__ whh,
                       const float* __restrict__ bih, const float* __restrict__ bhh) {
    int t = blockIdx.x * blockDim.x + threadIdx.x;
    if (t >= NB * DIM) return;
    int b = t >> 6, u = t & 63;
    const float* qb = q + (size_t)b * (2 * DIM);
    const float* hb = hin + (size_t)b * DIM;
    float g[4];
#pragma unroll
    for (int gg = 0; gg < 4; ++gg) {
        int rowi = gg * DIM + u;
        float acc = bih[rowi] + bhh[rowi];
        const float* wi = wih + (size_t)rowi * (2 * DIM);
        const float* wh = whh + (size_t)rowi * DIM;
        for (int k = 0; k < 2 * DIM; ++k) acc = fmaf(qb[k], wi[k], acc);
        for (int k = 0; k < DIM; ++k)     acc = fmaf(hb[k], wh[k], acc);
        g[gg] = acc;
    }
    float ii = sigmf(g[0]), ff = sigmf(g[1]), cg = tanhf(g[2]), oo = sigmf(g[3]);
    float cn = ff * c[t] + ii * cg;
    c[t] = cn;
    hout[t] = oo * tanhf(cn);
}

// e[n] = dot(feat[n], lh[graph_index[n]]) — wave per node, shfl reduction
__global__ void k_attn(const float* __restrict__ feat, const float* __restrict__ lh,
                       const int* __restrict__ gidx, float* __restrict__ ev, int N) {
    int w = blockIdx.x * (blockDim.x >> 5) + (threadIdx.x >> 5);
    if (w >= N) return;
    int lane = threadIdx.x & 31;
    int g = gidx[w];
    const float* f  = feat + (size_t)w * DIM;
    const float* hh = lh + (size_t)g * DIM;
    float s = f[lane] * hh[lane] + f[lane + 32] * hh[lane + 32];
#pragma unroll
    for (int d = 16; d >= 1; d >>= 1) s += __shfl_xor(s, d, 32);
    if (lane == 0) ev[w] = s;
}

// graph ranges from sorted graph_index
__global__ void k_bounds(const int* __restrict__ gidx, int* __restrict__ gs,
                         int* __restrict__ ge, int N) {
    int n = blockIdx.x * blockDim.x + threadIdx.x;
    if (n >= N) return;
    int g = gidx[n];
    if (n == 0 || gidx[n - 1] != g) gs[g] = n;
    if (n == N - 1 || gidx[n + 1] != g) ge[g] = n + 1;
}

// per-graph softmax stats (max, sum-exp) — wave per graph
__global__ void k_stats(const float* __restrict__ ev, const int* __restrict__ gs,
                        const int* __restrict__ ge, float* __restrict__ emax,
                        float* __restrict__ asum) {
    int w = blockIdx.x * (blockDim.x >> 5) + (threadIdx.x >> 5);
    if (w >= NB) return;
    int lane = threadIdx.x & 31;
    int s = gs[w], e = ge[w];
    float mx = -3.402823466e38f;
    for (int n = s + lane; n < e; n += 32) mx = fmaxf(mx, ev[n]);
#pragma unroll
    for (int d = 16; d >= 1; d >>= 1) mx = fmaxf(mx, __shfl_xor(mx, d, 32));
    float sm = 0.0f;
    for (int n = s + lane; n < e; n += 32) sm += expf(ev[n] - mx);
#pragma unroll
    for (int d = 16; d >= 1; d >>= 1) sm += __shfl_xor(sm, d, 32);
    if (lane == 0) { emax[w] = mx; asum[w] = (s < e) ? sm : 1.0f; }
}

// r[g,:] = sum_n a[n]*feat[n,:]; q_star[g] = [lh[g], r[g]] — wave per graph
__global__ void k_readout(const float* __restrict__ feat, const float* __restrict__ ev,
                          const int* __restrict__ gs, const int* __restrict__ ge,
                          const float* __restrict__ emax, const float* __restrict__ asum,
                          const float* __restrict__ lh, float* __restrict__ qstar) {
    int w = blockIdx.x * (blockDim.x >> 5) + (threadIdx.x >> 5);
    if (w >= NB) return;
    int lane = threadIdx.x & 31;
    int s = gs[w], e = ge[w];
    float mx = emax[w], inv = 1.0f / asum[w];
    float r0 = 0.0f, r1 = 0.0f;
    for (int n = s; n < e; ++n) {
        float a = expf(ev[n] - mx) * inv;
        const float* f = feat + (size_t)n * DIM;
        r0 = fmaf(a, f[lane], r0);
        r1 = fmaf(a, f[lane + 32], r1);
    }
    float* q = qstar + (size_t)w * (2 * DIM);
    q[lane]            = lh[(size_t)w * DIM + lane];
    q[lane + 32]       = lh[(size_t)w * DIM + lane + 32];
    q[DIM + lane]      = r0;
    q[DIM + lane + 32] = r1;
}

// pred[b] = y[b,:] @ fc2_w + fc2_b — wave per graph
__global__ void k_fc2(const float* __restrict__ y, const float* __restrict__ w,
                      const float* __restrict__ b, float* __restrict__ out) {
    int t = blockIdx.x * (blockDim.x >> 5) + (threadIdx.x >> 5);
    if (t >= NB) return;
    int lane = threadIdx.x & 31;
    const float* yb = y + (size_t)t * DIM;
    float s = yb[lane] * w[lane] + yb[lane + 32] * w[lane + 32];
#pragma unroll
    for (int d = 16; d >= 1; d >>= 1) s += __shfl_xor(s, d, 32);
    if (lane == 0) out[t] = s + b[0];
}

// ---------------------------------------------------------------------------
extern "C" void kernel_launch(void* const* d_in, const int* in_sizes, int n_in,
                              void* d_out, int out_size, void* d_ws, size_t ws_size,
                              hipStream_t stream) {
    (void)n_in; (void)out_size;
    const float* nf      = (const float*)d_in[0];
    const float* ef      = (const float*)d_in[1];
    const int*   eidx    = (const int*)d_in[2];
    const int*   gidx    = (const int*)d_in[3];
    const float* lin0_w  = (const float*)d_in[5];
    const float* lin0_b  = (const float*)d_in[6];
    const float* enn_w1  = (const float*)d_in[7];
    const float* enn_b1  = (const float*)d_in[8];
    const float* enn_w2  = (const float*)d_in[9];
    const float* enn_b2  = (const float*)d_in[10];
    const float* conv_b  = (const float*)d_in[11];
    const float* gru_wih = (const float*)d_in[12];
    const float* gru_whh = (const float*)d_in[13];
    const float* gru_bih = (const float*)d_in[14];
    const float* gru_bhh = (const float*)d_in[15];
    const float* s2s_wih = (const float*)d_in[16];
    const float* s2s_whh = (const float*)d_in[17];
    const float* s2s_bih = (const float*)d_in[18];
    const float* s2s_bhh = (const float*)d_in[19];
    const float* fc1_w   = (const float*)d_in[20];
    const float* fc1_b   = (const float*)d_in[21];
    const float* fc2_w   = (const float*)d_in[22];
    const float* fc2_b   = (const float*)d_in[23];

    const int N = in_sizes[0] / F_IN;
    const int E = in_sizes[1] / EDIM;
    const int* src = eidx;
    const int* dst = eidx + E;
    float* out = (float*)d_out;

    // ---- workspace carve-up (256B aligned regions; remainder -> ew) ----
    char* ws = (char*)d_ws;
    size_t off = 0;
    auto take = [&](size_t bytes) -> char* {
        char* p = ws + off; off = (off + bytes + 255) & ~(size_t)255; return p;
    };
    float* hstate = (float*)take((size_t)N * DIM * 4);
    float* agg    = (float*)take((size_t)N * DIM * 4);
    float* mbuf   = (float*)take((size_t)N * DIM * 4);
    float* Gi     = (float*)take((size_t)N * G3 * 4);
    float* Gh     = (float*)take((size_t)N * G3 * 4);
    float* h1     = (float*)take((size_t)E * HID * 4);
    float* invd   = (float*)take((size_t)N * 4);
    float* apad   = (float*)take((size_t)N * F_INP * 4);
    float* w2p    = (float*)take((size_t)HID * EWN * 4);     // packed enn_w2
    float* w0p    = (float*)take((size_t)F_INP * DIM * 4);   // packed padded lin0_w
    float* wihP   = (float*)take((size_t)DIM * G3 * 4);      // packed gru_wih^T
    float* whhP   = (float*)take((size_t)DIM * G3 * 4);      // packed gru_whh^T
    float* fc1p   = (float*)take((size_t)2 * DIM * DIM * 4); // packed fc1_w
    float* ev     = (float*)take((size_t)N * 4);
    float* emax   = (float*)take((size_t)NB * 4);
    float* asum   = (float*)take((size_t)NB * 4);
    float* lhA    = (float*)take((size_t)NB * DIM * 4);
    float* lhB    = (float*)take((size_t)NB * DIM * 4);
    float* lc     = (float*)take((size_t)NB * DIM * 4);
    float* qstar  = (float*)take((size_t)NB * 2 * DIM * 4);
    float* ybuf   = (float*)take((size_t)NB * DIM * 4);
    int*   gs     = (int*)take((size_t)NB * 4);
    int*   ge     = (int*)take((size_t)NB * 4);
    float* ewbuf  = (float*)(ws + off);
    size_t ew_bytes = (ws_size > off) ? (ws_size - off) : 0;
    long cap = (long)(ew_bytes / ((size_t)EWN * 4));
    cap &= ~15L;
    if (cap < 16) cap = 16;
    if (cap > E)  cap = E;
    const bool full = (cap >= E);

    #define CDIV(a, b) (int)(((a) + (b) - 1) / (b))
    auto gemm = [&](int nt, bool relu, const float* A, const float* Bp, float* C,
                    const float* bias, int M, int Nc, int K, int lda, int ldc) {
        int waves = CDIV(M, 16) * (Nc / (nt * 16));
        dim3 grid(CDIV(waves, 8));
        if (nt == 8) {
            if (relu) k_gemm_wmma<8, true ><<<grid, 256, 0, stream>>>(A, Bp, C, bias, M, Nc, K, lda, ldc);
            else      k_gemm_wmma<8, false><<<grid, 256, 0, stream>>>(A, Bp, C, bias, M, Nc, K, lda, ldc);
        } else {
            if (relu) k_gemm_wmma<4, true ><<<grid, 256, 0, stream>>>(A, Bp, C, bias, M, Nc, K, lda, ldc);
            else      k_gemm_wmma<4, false><<<grid, 256, 0, stream>>>(A, Bp, C, bias, M, Nc, K, lda, ldc);
        }
    };

    // ---- one-time packing / degrees / edge-net hidden ----
    k_pad_nf<<<CDIV((long)N * F_INP, 256), 256, 0, stream>>>(nf, apad, N);
    k_packB <<<CDIV((long)HID * EWN, 256), 256, 0, stream>>>(enn_w2, w2p, HID, HID, EWN);
    k_packB <<<CDIV(F_INP * DIM, 256),     256, 0, stream>>>(lin0_w, w0p, F_IN, F_INP, DIM);
    k_packB <<<CDIV(2 * DIM * DIM, 256),   256, 0, stream>>>(fc1_w, fc1p, 2 * DIM, 2 * DIM, DIM);
    k_packBT<<<CDIV(G3 * DIM, 256),        256, 0, stream>>>(gru_wih, wihP, G3, DIM);
    k_packBT<<<CDIV(G3 * DIM, 256),        256, 0, stream>>>(gru_whh, whhP, G3, DIM);
    k_fill  <<<CDIV((long)N, 256),         256, 0, stream>>>(invd, 0.0f, N);
    k_deg   <<<CDIV((long)E, 256),         256, 0, stream>>>(dst, invd, E);
    k_invdeg<<<CDIV((long)N, 256),         256, 0, stream>>>(invd, N);
    k_enn_hidden<<<CDIV((long)E * HID, 256), 256, 0, stream>>>(ef, enn_w1, enn_b1, h1, E);

    // lin0 + relu -> initial node state (WMMA, K=32)
    gemm(4, true, apad, w0p, hstate, lin0_b, N, DIM, F_INP, F_INP, DIM);

    // edge-conditioned weights ew[e] = h1 @ W2 + b2 (big WMMA GEMM, K=128)
    if (full)
        gemm(8, false, h1, w2p, ewbuf, enn_b2, E, EWN, HID, HID, EWN);

    // ---- 3 message-passing + GRU iterations ----
    for (int it = 0; it < 3; ++it) {
        k_fill<<<CDIV((long)N * DIM, 256), 256, 0, stream>>>(agg, 0.0f, (long)N * DIM);
        if (full) {
            k_msg<<<CDIV((long)E, 8), 256, 0, stream>>>(hstate, ewbuf, src, dst, agg, 0, E);
        } else {
            for (int e0 = 0; e0 < E; e0 += (int)cap) {
                int cnt = (E - e0 < (int)cap) ? (E - e0) : (int)cap;
                gemm(8, false, h1 + (size_t)e0 * HID, w2p, ewbuf, enn_b2,
                     cnt, EWN, HID, HID, EWN);
                k_msg<<<CDIV((long)cnt, 8), 256, 0, stream>>>(hstate, ewbuf, src, dst, agg, e0, cnt);
            }
        }
        k_mrelu<<<CDIV((long)N * DIM, 256), 256, 0, stream>>>(agg, invd, conv_b, mbuf, N);
        gemm(4, false, mbuf,   wihP, Gi, gru_bih, N, G3, DIM, DIM, G3);   // x-side gates
        gemm(4, false, hstate, whhP, Gh, gru_bhh, N, G3, DIM, DIM, G3);   // h-side gates
        k_gru<<<CDIV((long)N * DIM, 256), 256, 0, stream>>>(Gi, Gh, hstate, N);
    }

    // ---- Set2Set readout (3 steps) ----
    k_fill<<<CDIV((long)NB * 2 * DIM, 256), 256, 0, stream>>>(qstar, 0.0f, (long)NB * 2 * DIM);
    k_fill<<<CDIV((long)NB * DIM, 256),     256, 0, stream>>>(lhA, 0.0f, (long)NB * DIM);
    k_fill<<<CDIV((long)NB * DIM, 256),     256, 0, stream>>>(lc, 0.0f, (long)NB * DIM);
    k_fill<<<1, 256, 0, stream>>>((float*)gs, 0.0f, NB);
    k_fill<<<1, 256, 0, stream>>>((float*)ge, 0.0f, NB);
    k_bounds<<<CDIV((long)N, 256), 256, 0, stream>>>(gidx, gs, ge, N);

    float* hc = lhA; float* hn = lhB;
    for (int s = 0; s < 3; ++s) {
        k_lstm<<<CDIV((long)NB * DIM, 256), 256, 0, stream>>>(
            qstar, hc, hn, lc, s2s_wih, s2s_whh, s2s_bih, s2s_bhh);
        k_attn<<<CDIV((long)N, 8), 256, 0, stream>>>(hstate, hn, gidx, ev, N);
        k_stats<<<CDIV((long)NB, 8), 256, 0, stream>>>(ev, gs, ge, emax, asum);
        k_readout<<<CDIV((long)NB, 8), 256, 0, stream>>>(hstate, ev, gs, ge, emax, asum, hn, qstar);
        float* tmp = hc; hc = hn; hn = tmp;
    }

    // ---- head: fc1 (WMMA) + fc2 ----
    gemm(4, true, qstar, fc1p, ybuf, fc1_b, NB, DIM, 2 * DIM, 2 * DIM, DIM);
    k_fc2<<<CDIV((long)NB, 8), 256, 0, stream>>>(ybuf, fc2_w, fc2_b, out);
}